// Rasterizer_83597243449490
// MI455X (gfx1250) — compile-verified
//
#include <hip/hip_runtime.h>
#include <math.h>

// ---------------------------------------------------------------------------
// Rasterizer for MI455X (gfx1250, wave32).
//
// Per-face preprocessing converts every face into 4 affine functions of
// (px,py): b0,b1,b2 (barycentrics pre-scaled by 1/area) and zp (depth plane).
// Rasterization evaluates, per 16-pixel group and per 4-face group,
// D = A(16x4 coeffs) x B(4x16 pixel basis) with one V_WMMA_F32_16X16X4_F32.
// Invalid faces store all-zero planes -> zp==0 fails zp > Z_CLIP everywhere.
//
// Hot loop tracks only (bestZ, bestF); barycentrics of the winner are
// re-evaluated once after the scan (lazy bary), and the 3-way b>=0 test is a
// single v_min3 + compare. Min-z with lowest-face-index tie-break matches the
// reference's chunked argmin + strict-< scan semantics.
// ---------------------------------------------------------------------------

typedef float v2f __attribute__((ext_vector_type(2)));
typedef float v8f __attribute__((ext_vector_type(8)));

#define NV_TOT   4000
#define NF       2048
#define NGROUP   (NF / 4)          // 512 face groups of 4
#define H_IMG    135
#define W_IMG    240
#define NPIX     (H_IMG * W_IMG)   // 32400
#define FX_C     1000.0f
#define FY_C     1000.0f
#define CX_C     960.0f
#define CY_C     540.0f
#define WFULL    1920.0f
#define HFULL    1080.0f
#define SFULL    1080.0f           // min(H_FULL, W_FULL)
#define SPIX     135.0f            // min(H, W)
#define EPS_C    1e-8f
#define ZCLIP    0.01f

// ---------------------------------------------------------------------------
// Kernel 1: per-face plane setup.
// Table layout: group g = 4 faces, 16 rows x 4 floats = 64 floats per group.
//   row (slot*4 + 0) = (b0x, b0y, b0c, 0)
//   row (slot*4 + 1) = (b1x, b1y, b1c, 0)
//   row (slot*4 + 2) = (b2x, b2y, b2c, 0)
//   row (slot*4 + 3) = (zx,  zy,  zc,  0)
// WMMA A-matrix feed: lane m reads floats 16m+{0,1} (low half, K=0,1) or 16m+{8,9}
// (high half, K=2,3) of the group -> one b64 load per lane per group.
// ---------------------------------------------------------------------------
__global__ void face_prep_kernel(const float* __restrict__ verts,
                                 const int* __restrict__ faces,
                                 float* __restrict__ tbl) {
  int f = blockIdx.x * blockDim.x + threadIdx.x;
  if (f >= NF) return;

  float x[3], y[3], z[3];
#pragma unroll
  for (int k = 0; k < 3; ++k) {
    int vi = faces[3 * f + k];
    float vx = verts[3 * vi + 0];
    float vy = verts[3 * vi + 1];
    float vz = verts[3 * vi + 2];
    float xs = FX_C * (-vx) / vz + CX_C;
    float ys = FY_C * (vy) / vz + CY_C;
    x[k] = -(2.0f * xs - WFULL) / SFULL;
    y[k] = -(2.0f * ys - HFULL) / SFULL;
    z[k] = vz;
  }

  float area = (x[1] - x[0]) * (y[2] - y[0]) - (y[1] - y[0]) * (x[2] - x[0]);
  bool ok = (fabsf(area) > EPS_C) && (z[0] > ZCLIP) && (z[1] > ZCLIP) && (z[2] > ZCLIP);
  float inv = ok ? (1.0f / area) : 0.0f;   // !ok -> all planes zero -> zp==0 -> rejected

  // w0 = edge(v1,v2), w1 = edge(v2,v0), w2 = edge(v0,v1)
  // edge(a,b): coeff_px = ay-by, coeff_py = bx-ax, const = (by-ay)*ax - (bx-ax)*ay
  float c0x = (y[1] - y[2]) * inv;
  float c0y = (x[2] - x[1]) * inv;
  float c0c = ((y[2] - y[1]) * x[1] - (x[2] - x[1]) * y[1]) * inv;

  float c1x = (y[2] - y[0]) * inv;
  float c1y = (x[0] - x[2]) * inv;
  float c1c = ((y[0] - y[2]) * x[2] - (x[0] - x[2]) * y[2]) * inv;

  float c2x = (y[0] - y[1]) * inv;
  float c2y = (x[1] - x[0]) * inv;
  float c2c = ((y[1] - y[0]) * x[0] - (x[1] - x[0]) * y[0]) * inv;

  float zx = c0x * z[0] + c1x * z[1] + c2x * z[2];
  float zy = c0y * z[0] + c1y * z[1] + c2y * z[2];
  float zc = c0c * z[0] + c1c * z[1] + c2c * z[2];

  int g = f >> 2;
  int slot = f & 3;
  float* r = tbl + (size_t)g * 64 + slot * 16;   // 4 rows x 4 floats
  r[0] = c0x;  r[1] = c0y;  r[2]  = c0c;  r[3]  = 0.0f;
  r[4] = c1x;  r[5] = c1y;  r[6]  = c1c;  r[7]  = 0.0f;
  r[8] = c2x;  r[9] = c2y;  r[10] = c2c;  r[11] = 0.0f;
  r[12] = zx;  r[13] = zy;  r[14] = zc;   r[15] = 0.0f;
}

// ---------------------------------------------------------------------------
// Kernel 2: WMMA raster loop. Each wave owns 16 pixels (both lane halves map
// to the same 16 pixel columns). Per face group: one f32 WMMA gives
// (b0,b1,b2,zp) for 4 faces x 16 pixels; each lane scans its 2 faces
// (low half: faces 4g,4g+1; high half: 4g+2,4g+3) keeping only (z, face id);
// a shfl_xor(16) merge resolves the halves. Winner bary is recomputed once.
// Control flow is uniform around the WMMA so EXEC is all ones (ISA §7.12).
// ---------------------------------------------------------------------------
__global__ void raster_kernel(const float* __restrict__ tbl,
                              int* __restrict__ out_face,
                              float* __restrict__ out_bary) {
  const int lane   = threadIdx.x & 31;
  const int waveId = (blockIdx.x * blockDim.x + threadIdx.x) >> 5;
  const int col16  = lane & 15;
  const bool hi    = lane >= 16;

  int p = waveId * 16 + col16;
  const bool active = (p < NPIX);
  int pc = active ? p : (NPIX - 1);          // clamp: keep EXEC full, skip store
  int row = pc / W_IMG;
  int col = pc - row * W_IMG;

  float px = -(2.0f * (float)col + 1.0f - (float)W_IMG) / SPIX;
  float py = -(2.0f * (float)row + 1.0f - (float)H_IMG) / SPIX;

  // B matrix (4x16): K=0 -> px, K=1 -> py, K=2 -> 1, K=3 -> 0.
  // VGPR0: lanes0-15 K=0, lanes16-31 K=2 ; VGPR1: K=1 / K=3.
  v2f B;
  B.x = hi ? 1.0f : px;
  B.y = hi ? 0.0f : py;

  const float INF = __builtin_inff();
  float bestZ = INF;
  int   bestF = -1;

  const int faceOff = hi ? 2 : 0;            // low half: faces 4g+{0,1}; high: 4g+{2,3}
  const float* aBase = tbl + (size_t)col16 * 16 + (hi ? 8 : 0);

  for (int g = 0; g < NGROUP; ++g) {
    // A matrix (16x4): lane m, low half reads group floats 16m+{0,1}, high 16m+{8,9}.
    const float* rp = aBase + (size_t)g * 64;
    v2f A;
    A.x = rp[0];
    A.y = rp[1];

    v8f c = {};
    v8f d = __builtin_amdgcn_wmma_f32_16x16x4_f32(
        /*neg_a=*/false, A, /*neg_b=*/false, B,
        /*c_mod=*/(short)0, c, /*reuse_a=*/false, /*reuse_b=*/false);

    int fid = g * 4 + faceOff;

    // face (rows 0..3 of this lane's half): d[0..3] = b0,b1,b2,zp
    {
      float mn = fminf(fminf(d[0], d[1]), d[2]);   // -> v_min3_num_f32
      float zp = d[3];
      bool upd = (mn >= 0.0f) & (zp > ZCLIP) & (zp < bestZ);  // strict <: low idx wins ties
      bestZ = upd ? zp : bestZ;
      bestF = upd ? fid : bestF;
    }
    // next face (rows 4..7): d[4..7]
    {
      float mn = fminf(fminf(d[4], d[5]), d[6]);
      float zp = d[7];
      bool upd = (mn >= 0.0f) & (zp > ZCLIP) & (zp < bestZ);
      bestZ = upd ? zp : bestZ;
      bestF = upd ? (fid + 1) : bestF;
    }
  }

  // Merge lane L (faces ==0,1 mod 4) with lane L+16 (faces ==2,3 mod 4):
  // min z, ties -> lowest face index (global reference semantics).
  float oz = __shfl_xor(bestZ, 16);
  int   of = __shfl_xor(bestF, 16);
  bool take = (oz < bestZ) || ((oz == bestZ) && (of < bestF));
  bestZ = take ? oz : bestZ;
  bestF = take ? of : bestF;

  if (!hi && active) {
    // Lazy barycentrics: re-evaluate the winning face's planes (6 FMAs).
    float bb0 = -1.0f, bb1 = -1.0f, bb2 = -1.0f;
    if (bestF >= 0) {
      const float* r = tbl + (size_t)(bestF >> 2) * 64 + (bestF & 3) * 16;
      bb0 = fmaf(r[0], px, fmaf(r[1], py, r[2]));
      bb1 = fmaf(r[4], px, fmaf(r[5], py, r[6]));
      bb2 = fmaf(r[8], px, fmaf(r[9], py, r[10]));
    }
    int orow = (H_IMG - 1) - row;            // reference flips rows ([::-1])
    int idx = orow * W_IMG + col;
    out_face[idx] = bestF;
    out_bary[idx * 3 + 0] = bb0;
    out_bary[idx * 3 + 1] = bb1;
    out_bary[idx * 3 + 2] = bb2;
  }
}

// ---------------------------------------------------------------------------
// d_in[0]: vertices float32 [4000,3]; d_in[1]: faces int32 [2048,3].
// d_out: pix_to_face int32 [135,240] (written via int* reinterpret) followed
// by bary_coords float32 [135,240,3]. d_ws: 512 groups * 256 B = 128 KB table.
// ---------------------------------------------------------------------------
extern "C" void kernel_launch(void* const* d_in, const int* in_sizes, int n_in,
                              void* d_out, int out_size, void* d_ws, size_t ws_size,
                              hipStream_t stream) {
  const float* verts = (const float*)d_in[0];
  const int* faces = (const int*)d_in[1];
  float* tbl = (float*)d_ws;
  int* out_face = (int*)d_out;
  float* out_bary = (float*)d_out + NPIX;

  face_prep_kernel<<<(NF + 255) / 256, 256, 0, stream>>>(verts, faces, tbl);

  // 16 pixels per wave, 8 waves per 256-thread block -> 128 pixels per block.
  int blocks = (NPIX + 127) / 128;
  raster_kernel<<<blocks, 256, 0, stream>>>(tbl, out_face, out_bary);
}